// FeatureCrossAttention_81260781240506
// MI455X (gfx1250) — compile-verified
//
#include <hip/hip_runtime.h>
#include <hip/hip_bf16.h>
#include <stdint.h>

// FeatureCrossAttention, algebraically reduced:
//   scores[f,g] = t_f * x_g + const_f   (row-constants cancel in softmax)
//   t_f = (A*x_f + C)/4,  A = wq.wk, C = bq.wk
//   out[f] = wv * (sum_g e_fg * x_g)/(sum_g e_fg) + bv,  e_fg = exp(t_f*x_g)
// => no matmul remains; the operator is transcendental-bound (67M v_exp_f32
//    ~3.3us vs 0.7us HBM at 23.3 TB/s). Data is streamed through LDS with
//    CDNA5 async global->LDS copies (ASYNCcnt), double buffered, one wave32
//    per (b,l) position, lane f owns output feature f.

#define CHUNK_POS       8      // positions per chunk == waves per block
#define CHUNK_FLOATS    256    // 8 rows * 32 floats
#define CPB             8      // chunks per block

__global__ __launch_bounds__(256, 4)
void fca_kernel(const float* __restrict__ x,
                const float* __restrict__ wq, const float* __restrict__ bq,
                const float* __restrict__ wk, const float* __restrict__ bk,
                const float* __restrict__ wv, const float* __restrict__ bv,
                float* __restrict__ out, int P, int total, int totalChunks)
{
    __shared__ __align__(16) float smem[2 * CHUNK_FLOATS];  // double buffer, 2KB

    const int t    = threadIdx.x;
    const int w    = t >> 5;
    const int lane = t & 31;

    // Uniform weight preprocessing -> compiler emits scalar-float ops (s_fmaak_f32).
    float A = 0.f, C = 0.f;
#pragma unroll
    for (int d = 0; d < 16; ++d) {
        A = fmaf(wq[d], wk[d], A);
        C = fmaf(bq[d], wk[d], C);
    }
    const float wv0 = wv[0], bv0 = bv[0];
    const float LOG2E = 1.4426950408889634f;

    // LDS byte address: low 32 bits of the generic (flat) address of a
    // __shared__ object are the DS offset (LDS aperture = {SHARED_BASE, off}).
    const unsigned lds_base = (unsigned)(unsigned long long)(uintptr_t)smem;

    const int c0 = blockIdx.x * CPB;
    if (c0 >= totalChunks) return;
    int cEnd = c0 + CPB;
    if (cEnd > totalChunks) cEnd = totalChunks;

    const unsigned utotal = (unsigned)total;

    // Lane l of wave w copies row w's element l -> each wave copies exactly
    // its own row: barrier-free double buffering, each wave waits only on its
    // own ASYNCcnt.
    auto issue_copy = [&](int c, int buf) {
        unsigned idx = (unsigned)c * CHUNK_FLOATS + (unsigned)t;
        if (idx >= utotal) idx = utotal - 1;               // tail clamp
        unsigned long long ga = (unsigned long long)(uintptr_t)(x + idx);
        unsigned la = lds_base + (unsigned)(buf * (CHUNK_FLOATS * 4) + t * 4);
        asm volatile("global_load_async_to_lds_b32 %0, %1, off"
                     :: "v"(la), "v"(ga) : "memory");
    };

    issue_copy(c0, 0);                                      // prologue

    for (int c = c0; c < cEnd; ++c) {
        const int buf = (c - c0) & 1;
        const int cn  = (c + 1 < totalChunks) ? (c + 1) : c;
        issue_copy(cn, buf ^ 1);                            // prefetch next chunk
        asm volatile("s_wait_asynccnt 1" ::: "memory");     // current chunk landed

        const int rbase = buf * CHUNK_FLOATS + w * 32;

        // Whole row into VGPRs (vectorizes to ds_load_b128 x8, broadcast reads).
        float xr[32];
#pragma unroll
        for (int g = 0; g < 32; ++g) xr[g] = smem[rbase + g];
        // Per-lane own value x_f (stride-4B -> bank-conflict-free ds_load_b32).
        const float xf = smem[rbase + lane];

        // Wave-wide min/max of the row: O(1) softmax stabilization,
        // row-max of em*x_g == max(em*xmax, em*xmin).
        float xmax = xf, xmin = xf;
#pragma unroll
        for (int off = 16; off > 0; off >>= 1) {
            xmax = fmaxf(xmax, __shfl_xor(xmax, off, 32));
            xmin = fminf(xmin, __shfl_xor(xmin, off, 32));
        }

        const float tf    = fmaf(A, xf, C) * 0.25f;
        const float em    = tf * LOG2E;                       // exp2 domain
        const float mterm = fmaxf(em * xmax, em * xmin);

        // Inner loop: 1 fma + 1 v_exp_f32 + 2 accum ops per score element.
        // Two interleaved partial accumulators halve the serial-add chain so
        // VALU latency stays hidden under the TRANS (exp) pipe.
        float s0a = 0.f, s0b = 0.f, s1a = 0.f, s1b = 0.f;
#pragma unroll
        for (int g = 0; g < 32; g += 2) {
            const float pa = fmaf(em, xr[g],     -mterm);
            const float pb = fmaf(em, xr[g + 1], -mterm);
            const float ea = __builtin_amdgcn_exp2f(pa);      // raw v_exp_f32
            const float eb = __builtin_amdgcn_exp2f(pb);
            s0a += ea;
            s0b += eb;
            s1a = fmaf(xr[g],     ea, s1a);
            s1b = fmaf(xr[g + 1], eb, s1b);
        }
        const float s0 = s0a + s0b;
        const float s1 = s1a + s1b;

        const int pos = c * CHUNK_POS + w;
        if (pos < P) {
            const float r = s1 * __builtin_amdgcn_rcpf(s0);   // v_rcp_f32
            out[(long)pos * 32 + lane] = fmaf(wv0, r, bv0);
        }
    }
    asm volatile("s_wait_asynccnt 0" ::: "memory");           // drain before end
}

extern "C" void kernel_launch(void* const* d_in, const int* in_sizes, int n_in,
                              void* d_out, int out_size, void* d_ws, size_t ws_size,
                              hipStream_t stream)
{
    const float* x  = (const float*)d_in[0];
    const float* wq = (const float*)d_in[1];
    const float* bq = (const float*)d_in[2];
    const float* wk = (const float*)d_in[3];
    const float* bk = (const float*)d_in[4];
    const float* wv = (const float*)d_in[5];
    const float* bv = (const float*)d_in[6];
    float* out = (float*)d_out;

    const int total       = in_sizes[0];       // B*L*F elements of x
    const int P           = total / 32;        // B*L positions (F = 32)
    const int totalChunks = (P + CHUNK_POS - 1) / CHUNK_POS;
    const int blocks      = (totalChunks + CPB - 1) / CPB;   // 1024 for 16x4096

    fca_kernel<<<blocks, 256, 0, stream>>>(x, wq, bq, wk, bk, wv, bv,
                                           out, P, total, totalChunks);
}